// PPOTrustGNN_46351287058836
// MI455X (gfx1250) — compile-verified
//
#include <hip/hip_runtime.h>
#include <math.h>

// ---------------------------------------------------------------------------
// MI455X / gfx1250 implementation of the PPO hetero-GAT forward pass.
// f32 end-to-end; dense projections via V_WMMA_F32_16X16X4_F32 with a
// 16x(16*NTILES) wave tile (A-fragment reuse across NTILES WMMAs).
// ---------------------------------------------------------------------------

typedef __attribute__((ext_vector_type(2))) float v2f;
typedef __attribute__((ext_vector_type(8))) float v8f;

#if defined(__HIP_DEVICE_COMPILE__) && !__has_builtin(__builtin_amdgcn_wmma_f32_16x16x4_f32)
#error "wmma f32 16x16x4 builtin not available on this toolchain"
#endif

__device__ __forceinline__ void atomicMaxF(float* addr, float val) {
  int* ai = (int*)addr;
  int cur = __float_as_int(*addr);
  while (val > __int_as_float(cur)) {
    int assumed = cur;
    cur = atomicCAS(ai, assumed, __float_as_int(val));
    if (cur == assumed) break;
  }
}

// ---------------- generic fill ----------------
__global__ void k_fill(float* __restrict__ p, float v, int n) {
  int i = blockIdx.x * blockDim.x + threadIdx.x;
  if (i < n) p[i] = v;
}

// ---------------- WMMA f32 GEMM: C[M,N] = act(A[M,K] @ B[K,N] + bias) ------
// Requires M%16==0, K%4==0, N%(16*NTILES)==0 (true for all call sites).
// One wave computes a 16 x (16*NTILES) strip: a single A fragment feeds
// NTILES consecutive v_wmma_f32_16x16x4_f32 issues (4x less A traffic).
// A layout (16x4 f32): lanes 0-15 = rows, VGPR0/1 = K{0,1} (lanes<16) or
// K{2,3} (lanes>=16).  B layout (4x16): mirrored on the K axis.
template <int NTILES>
__global__ void k_wmma_gemm(const float* __restrict__ A, const float* __restrict__ B,
                            const float* __restrict__ bias, float* __restrict__ C,
                            int M, int K, int N, int doRelu) {
  int lane = threadIdx.x & 31;
  int wave = threadIdx.x >> 5;
  int m0 = (blockIdx.x * 4 + wave) * 16;
  int n0 = blockIdx.y * (16 * NTILES);
  if (m0 >= M) return;                 // wave-uniform: EXEC stays all-ones
  int half = lane >> 4;                // 0 or 1
  int l16  = lane & 15;
  int kb   = half * 2;

  v8f c[NTILES];
#pragma unroll
  for (int i = 0; i < NTILES; ++i) c[i] = (v8f){};

  for (int k = 0; k < K; k += 4) {
    const float* ap = A + (size_t)(m0 + l16) * K + (k + kb);
    v2f a; a.x = ap[0]; a.y = ap[1];
    const float* bp = B + (size_t)(k + kb) * N + n0 + l16;
#pragma unroll
    for (int i = 0; i < NTILES; ++i) {
      v2f b; b.x = bp[i * 16]; b.y = bp[N + i * 16];
      c[i] = __builtin_amdgcn_wmma_f32_16x16x4_f32(false, a, false, b,
                                                   (short)0, c[i], false, false);
    }
  }
#pragma unroll
  for (int i = 0; i < NTILES; ++i) {
    int col = n0 + i * 16 + l16;
    float bv = bias ? bias[col] : 0.0f;
#pragma unroll
    for (int r = 0; r < 8; ++r) {
      int row = m0 + r + half * 8;
      float v = c[i][r] + bv;
      if (doRelu) v = fmaxf(v, 0.0f);
      C[(size_t)row * N + col] = v;
    }
  }
}

// ---------------- attention-vector projection: proj[k][h] = sum_c W[k][h*64+c]*a[h][c]
__global__ void k_proj(const float* __restrict__ W, const float* __restrict__ asrc,
                       const float* __restrict__ adst,
                       float* __restrict__ ps, float* __restrict__ pd) {
  int t = threadIdx.x;            // 256 threads
  int k = t >> 2, h = t & 3;
  const float* wr = W + (size_t)k * 256 + h * 64;
  const float* as = asrc + h * 64;
  const float* ad = adst + h * 64;
  float s = 0.f, d = 0.f;
#pragma unroll 8
  for (int c = 0; c < 64; ++c) { s += wr[c] * as[c]; d += wr[c] * ad[c]; }
  ps[k * 4 + h] = s;
  pd[k * 4 + h] = d;
}

// ---------------- per-node attention logits: al[n][h] = x[n,:] . proj[:,h]
__global__ void k_al(const float* __restrict__ x, const float* __restrict__ proj,
                     float* __restrict__ al, int N) {
  int t = blockIdx.x * blockDim.x + threadIdx.x;
  if (t >= N * 4) return;
  int n = t >> 2, h = t & 3;
  const float* xr = x + (size_t)n * 64;
  float s = 0.f;
#pragma unroll 8
  for (int k = 0; k < 64; ++k) s += xr[k] * proj[k * 4 + h];
  al[t] = s;
}

__device__ __forceinline__ float leaky02(float v) { return v > 0.f ? v : 0.2f * v; }

// ---------------- edge segment-max ----------------
__global__ void k_edge_max(const int* __restrict__ src, const int* __restrict__ dst,
                           const float* __restrict__ als, const float* __restrict__ ald,
                           float* __restrict__ m, int E) {
  int t = blockIdx.x * blockDim.x + threadIdx.x;
  if (t >= E * 4) return;
  int e = t >> 2, h = t & 3;
  int d = dst[e];
  float v = leaky02(als[src[e] * 4 + h] + ald[d * 4 + h]);
  atomicMaxF(&m[d * 4 + h], v);
}

__global__ void k_fixm(float* __restrict__ m, int n) {
  int i = blockIdx.x * blockDim.x + threadIdx.x;
  if (i < n) { float v = m[i]; if (v < -1e30f) m[i] = 0.f; }
}

// ---------------- exp + segment-sum ----------------
__global__ void k_edge_expsum(const int* __restrict__ src, const int* __restrict__ dst,
                              const float* __restrict__ als, const float* __restrict__ ald,
                              const float* __restrict__ m, float* __restrict__ ex,
                              float* __restrict__ den, int E) {
  int t = blockIdx.x * blockDim.x + threadIdx.x;
  if (t >= E * 4) return;
  int e = t >> 2, h = t & 3;
  int d = dst[e];
  float v = leaky02(als[src[e] * 4 + h] + ald[d * 4 + h]);
  float x = expf(v - m[d * 4 + h]);
  ex[t] = x;
  atomicAdd(&den[d * 4 + h], x);
}

__global__ void k_invden(float* __restrict__ den, int n) {
  int i = blockIdx.x * blockDim.x + threadIdx.x;
  if (i < n) den[i] = 1.0f / (den[i] + 1e-16f);
}

__global__ void k_alpha(const int* __restrict__ dst, float* __restrict__ ex,
                        const float* __restrict__ invden, int E) {
  int t = blockIdx.x * blockDim.x + threadIdx.x;
  if (t >= E * 4) return;
  int e = t >> 2, h = t & 3;
  ex[t] *= invden[dst[e] * 4 + h];
}

// ---------------- weighted scatter-add: acc[dst][h*64+c] += alpha*h_src[src][h*64+c]
__global__ void k_aggr(const int* __restrict__ src, const int* __restrict__ dst,
                       const float* __restrict__ alpha, const float* __restrict__ hsrc,
                       float* __restrict__ acc, int E) {
  int t = blockIdx.x * blockDim.x + threadIdx.x;
  if (t >= E * 64) return;
  int e = t >> 6, c0 = t & 63;
  int s = src[e], d = dst[e];
  if (e + 16 < E)  // warm L0/L2 for an upcoming edge's source row
    __builtin_prefetch(hsrc + (size_t)src[e + 16] * 256 + c0, 0, 0);
  const float* hr = hsrc + (size_t)s * 256 + c0;
  float* ar = acc + (size_t)d * 256 + c0;
  const float* al = alpha + (size_t)e * 4;
#pragma unroll
  for (int h = 0; h < 4; ++h)
    atomicAdd(ar + h * 64, al[h] * hr[h * 64]);
}

// ---------------- head-mean + biases + relu, with LDS-staged BN statistics --
__global__ void k_finalize(const float* __restrict__ acc, const float* __restrict__ b0,
                           const float* __restrict__ b2, float* __restrict__ tmp,
                           float* __restrict__ csum, float* __restrict__ csq, int N) {
  __shared__ float ssum[64], ssq[64];
  int t = threadIdx.x;                 // 256
  if (t < 64) { ssum[t] = 0.f; ssq[t] = 0.f; }
  __syncthreads();
  int gi = blockIdx.x * 256 + t;
  if (gi < N * 64) {
    int n = gi >> 6, c = gi & 63;
    const float* ar = acc + (size_t)n * 256 + c;
    float v = 0.125f * (ar[0] + ar[64] + ar[128] + ar[192]) + 0.5f * (b0[c] + b2[c]);
    v = fmaxf(v, 0.f);
    tmp[gi] = v;
    atomicAdd(&ssum[c], v);            // ds_add_f32
    atomicAdd(&ssq[c], v * v);
  }
  __syncthreads();
  if (t < 64) { atomicAdd(&csum[t], ssum[t]); atomicAdd(&csq[t], ssq[t]); }
}

// ---------------- batchnorm apply (+ optional residual) ----------------
__global__ void k_bn(const float* __restrict__ tmp, const float* __restrict__ csum,
                     const float* __restrict__ csq, const float* __restrict__ g,
                     const float* __restrict__ b, const float* __restrict__ prev,
                     float* __restrict__ out, int N) {
  int t = blockIdx.x * blockDim.x + threadIdx.x;
  if (t >= N * 64) return;
  int c = t & 63;
  float inv = 1.0f / (float)N;
  float mu = csum[c] * inv;
  float var = csq[c] * inv - mu * mu;
  float y = g[c] * (tmp[t] - mu) * rsqrtf(var + 1e-5f) + b[c];
  if (prev) y += prev[t];
  out[t] = y;
}

// ---------------- actor head tail: LN(32) -> relu -> dot W2 -> softplus+1 ---
__global__ void k_head(const float* __restrict__ h1, const float* __restrict__ g,
                       const float* __restrict__ be, const float* __restrict__ W2,
                       const float* __restrict__ b2, float* __restrict__ out, int N) {
  int n = blockIdx.x * blockDim.x + threadIdx.x;
  if (n >= N) return;
  const float* hr = h1 + (size_t)n * 32;
  float mu = 0.f;
#pragma unroll
  for (int i = 0; i < 32; ++i) mu += hr[i];
  mu *= (1.0f / 32.0f);
  float var = 0.f;
#pragma unroll
  for (int i = 0; i < 32; ++i) { float d = hr[i] - mu; var += d * d; }
  var *= (1.0f / 32.0f);
  float r = rsqrtf(var + 1e-5f);
  float s = b2[0];
#pragma unroll
  for (int i = 0; i < 32; ++i) {
    float y = fmaxf(g[i] * (hr[i] - mu) * r + be[i], 0.f);
    s += y * W2[i];
  }
  float sp = (s > 20.f) ? s : log1pf(expf(s));
  out[n] = sp + 1.0f;
}

__global__ void k_ratio(const float* __restrict__ a, const float* __restrict__ b,
                        float* __restrict__ o, int N) {
  int i = blockIdx.x * blockDim.x + threadIdx.x;
  if (i < N) o[i] = a[i] / (a[i] + b[i]);
}

// ---------------- per-column sum / sumsq / max (LDS two-level) ----------------
__global__ void k_colstats(const float* __restrict__ x, float* __restrict__ sum,
                           float* __restrict__ sq, float* __restrict__ mx, int N) {
  __shared__ float ss[64], sqs[64], sm[64];
  int t = threadIdx.x;
  if (t < 64) { ss[t] = 0.f; sqs[t] = 0.f; sm[t] = -3e38f; }
  __syncthreads();
  int gi = blockIdx.x * 256 + t;
  if (gi < N * 64) {
    float v = x[gi];
    int c = gi & 63;
    atomicAdd(&ss[c], v);
    atomicAdd(&sqs[c], v * v);
    atomicMaxF(&sm[c], v);
  }
  __syncthreads();
  if (t < 64) {
    atomicAdd(&sum[t], ss[t]);
    atomicAdd(&sq[t], sqs[t]);
    atomicMaxF(&mx[t], sm[t]);
  }
}

// ---------------- critic MLP, one block of 128 threads ----------------
__global__ void k_critic(const float* __restrict__ sa, const float* __restrict__ qa,
                         const float* __restrict__ ma, const float* __restrict__ st,
                         const float* __restrict__ qt, const float* __restrict__ mt,
                         const float* __restrict__ cW1, const float* __restrict__ cb1,
                         const float* __restrict__ cg1, const float* __restrict__ cbe1,
                         const float* __restrict__ cW2, const float* __restrict__ cb2,
                         const float* __restrict__ cg2, const float* __restrict__ cbe2,
                         const float* __restrict__ cW3, const float* __restrict__ cb3,
                         const float* __restrict__ cW4, const float* __restrict__ cb4,
                         float* __restrict__ out, int NAn, int NTn) {
  __shared__ float g[384];
  __shared__ float h1[128];
  __shared__ float h2[64];
  __shared__ float h3[32];
  __shared__ float red[2];
  int t = threadIdx.x;  // 128
  if (t < 64) {
    float muA = sa[t] / (float)NAn;
    float vA = (qa[t] - (float)NAn * muA * muA) / (float)(NAn - 1);
    float muT = st[t] / (float)NTn;
    float vT = (qt[t] - (float)NTn * muT * muT) / (float)(NTn - 1);
    g[t]       = muA;
    g[64 + t]  = ma[t];
    g[128 + t] = sqrtf(fmaxf(vA, 0.f));
    g[192 + t] = muT;
    g[256 + t] = mt[t];
    g[320 + t] = sqrtf(fmaxf(vT, 0.f));
  }
  __syncthreads();
  // ---- h1 = relu(LN(g @ cW1 + cb1)), 384 -> 128
  float s1 = cb1[t];
  for (int k = 0; k < 384; ++k) s1 += g[k] * cW1[k * 128 + t];
  if (t == 0) { red[0] = 0.f; red[1] = 0.f; }
  __syncthreads();
  atomicAdd(&red[0], s1);
  atomicAdd(&red[1], s1 * s1);
  __syncthreads();
  {
    float mu = red[0] * (1.0f / 128.0f);
    float var = red[1] * (1.0f / 128.0f) - mu * mu;
    h1[t] = fmaxf(cg1[t] * (s1 - mu) * rsqrtf(var + 1e-5f) + cbe1[t], 0.f);
  }
  __syncthreads();
  // ---- h2 = relu(LN(h1 @ cW2 + cb2)), 128 -> 64
  float s2 = 0.f;
  if (t < 64) {
    s2 = cb2[t];
    for (int k = 0; k < 128; ++k) s2 += h1[k] * cW2[k * 64 + t];
  }
  if (t == 0) { red[0] = 0.f; red[1] = 0.f; }
  __syncthreads();
  if (t < 64) { atomicAdd(&red[0], s2); atomicAdd(&red[1], s2 * s2); }
  __syncthreads();
  if (t < 64) {
    float mu = red[0] * (1.0f / 64.0f);
    float var = red[1] * (1.0f / 64.0f) - mu * mu;
    h2[t] = fmaxf(cg2[t] * (s2 - mu) * rsqrtf(var + 1e-5f) + cbe2[t], 0.f);
  }
  __syncthreads();
  // ---- h3 = relu(h2 @ cW3 + cb3), 64 -> 32
  if (t < 32) {
    float s3 = cb3[t];
    for (int k = 0; k < 64; ++k) s3 += h2[k] * cW3[k * 32 + t];
    h3[t] = fmaxf(s3, 0.f);
  }
  __syncthreads();
  if (t == 0) {
    float s4 = cb4[0];
    for (int k = 0; k < 32; ++k) s4 += h3[k] * cW4[k];
    out[0] = s4;
  }
}

// ---------------------------------------------------------------------------
extern "C" void kernel_launch(void* const* d_in, const int* in_sizes, int n_in,
                              void* d_out, int out_size, void* d_ws, size_t ws_size,
                              hipStream_t stream) {
  const int NA = in_sizes[0] / 32;   // 8000
  const int NT = in_sizes[1] / 48;   // 30000
  const int E  = in_sizes[2];        // 150000

  const float* x_agent = (const float*)d_in[0];
  const float* x_track = (const float*)d_in[1];
  const int* srcs[4] = {(const int*)d_in[2], (const int*)d_in[4],
                        (const int*)d_in[6], (const int*)d_in[8]};
  const int* dsts[4] = {(const int*)d_in[3], (const int*)d_in[5],
                        (const int*)d_in[7], (const int*)d_in[9]};
  const float* ae_W = (const float*)d_in[10];
  const float* ae_b = (const float*)d_in[11];
  const float* te_W = (const float*)d_in[12];
  const float* te_b = (const float*)d_in[13];
  const float* conv_W    = (const float*)d_in[14];
  const float* conv_asrc = (const float*)d_in[15];
  const float* conv_adst = (const float*)d_in[16];
  const float* conv_bias = (const float*)d_in[17];
  const float* bn_gamma  = (const float*)d_in[18];
  const float* bn_beta   = (const float*)d_in[19];
  const float* aW1 = (const float*)d_in[20];
  const float* ab1 = (const float*)d_in[21];
  const float* ag  = (const float*)d_in[22];
  const float* abe = (const float*)d_in[23];
  const float* aW2 = (const float*)d_in[24];
  const float* ab2 = (const float*)d_in[25];
  const float* cW1 = (const float*)d_in[26];
  const float* cb1 = (const float*)d_in[27];
  const float* cg1 = (const float*)d_in[28];
  const float* cbe1 = (const float*)d_in[29];
  const float* cW2 = (const float*)d_in[30];
  const float* cb2 = (const float*)d_in[31];
  const float* cg2 = (const float*)d_in[32];
  const float* cbe2 = (const float*)d_in[33];
  const float* cW3 = (const float*)d_in[34];
  const float* cb3 = (const float*)d_in[35];
  const float* cW4 = (const float*)d_in[36];
  const float* cb4 = (const float*)d_in[37];

  // ---- workspace carve (floats) ----
  float* w = (float*)d_ws;
  float* xa   = w; w += (size_t)NA * 64;
  float* xt   = w; w += (size_t)NT * 64;
  float* nxa  = w; w += (size_t)NA * 64;
  float* nxt  = w; w += (size_t)NT * 64;
  float* hsrc = w; w += (size_t)NT * 256;   // also reused as BN tmp buffer
  float* acct = w; w += (size_t)NT * 256;
  float* acca = w; w += (size_t)NA * 256;
  float* als  = w; w += (size_t)NT * 4;
  float* ald  = w; w += (size_t)NT * 4;
  float* mbuf = w; w += (size_t)NT * 4;
  float* den  = w; w += (size_t)NT * 4;
  float* ex   = w; w += (size_t)E * 4;
  float* ps   = w; w += 256;
  float* pd   = w; w += 256;
  float* csum = w; w += 64;
  float* csq  = w; w += 64;
  float* h1b  = w; w += (size_t)NT * 32;
  float* cstat = w; w += 64 * 6;            // sa qa ma st qt mt

  auto fill = [&](float* p, float v, int n) {
    k_fill<<<(n + 255) / 256, 256, 0, stream>>>(p, v, n);
  };
  auto gemm = [&](const float* A, const float* B, const float* bias, float* Cc,
                  int M, int K, int N, int relu) {
    if (N % 64 == 0) {
      dim3 grid((M + 63) / 64, N / 64);
      k_wmma_gemm<4><<<grid, 128, 0, stream>>>(A, B, bias, Cc, M, K, N, relu);
    } else {  // N == 32 (actor-head hidden)
      dim3 grid((M + 63) / 64, N / 32);
      k_wmma_gemm<2><<<grid, 128, 0, stream>>>(A, B, bias, Cc, M, K, N, relu);
    }
  };

  // ---- node encoders ----
  gemm(x_agent, ae_W, ae_b, xa, NA, 32, 64, 1);
  gemm(x_track, te_W, te_b, xt, NT, 48, 64, 1);

  auto do_rel = [&](int l, int r, const float* xsrc, int Ns, const float* xdst,
                    int Nd, float* acc) {
    const float* Wlr = conv_W + (size_t)(l * 4 + r) * 64 * 256;
    const float* asr = conv_asrc + (size_t)(l * 4 + r) * 256;
    const float* adl = conv_adst + (size_t)(l * 4 + r) * 256;
    const int* sI = srcs[r];
    const int* dI = dsts[r];
    gemm(xsrc, Wlr, nullptr, hsrc, Ns, 64, 256, 0);                 // h_src
    k_proj<<<1, 256, 0, stream>>>(Wlr, asr, adl, ps, pd);
    k_al<<<(Ns * 4 + 255) / 256, 256, 0, stream>>>(xsrc, ps, als, Ns);
    k_al<<<(Nd * 4 + 255) / 256, 256, 0, stream>>>(xdst, pd, ald, Nd);
    fill(mbuf, -3e38f, Nd * 4);
    k_edge_max<<<(E * 4 + 255) / 256, 256, 0, stream>>>(sI, dI, als, ald, mbuf, E);
    k_fixm<<<(Nd * 4 + 255) / 256, 256, 0, stream>>>(mbuf, Nd * 4);
    fill(den, 0.f, Nd * 4);
    k_edge_expsum<<<(E * 4 + 255) / 256, 256, 0, stream>>>(sI, dI, als, ald, mbuf, ex, den, E);
    k_invden<<<(Nd * 4 + 255) / 256, 256, 0, stream>>>(den, Nd * 4);
    k_alpha<<<(E * 4 + 255) / 256, 256, 0, stream>>>(dI, ex, den, E);
    k_aggr<<<(E * 64 + 255) / 256, 256, 0, stream>>>(sI, dI, ex, hsrc, acc, E);
  };

  float* cxa = xa; float* cxt = xt; float* oxa = nxa; float* oxt = nxt;
  for (int l = 0; l < 3; ++l) {
    // track-target relations (0, 2): src=agent, dst=track
    fill(acct, 0.f, NT * 256);
    do_rel(l, 0, cxa, NA, cxt, NT, acct);
    do_rel(l, 2, cxa, NA, cxt, NT, acct);
    fill(csum, 0.f, 64); fill(csq, 0.f, 64);
    k_finalize<<<(NT * 64 + 255) / 256, 256, 0, stream>>>(
        acct, conv_bias + (size_t)(l * 4 + 0) * 64, conv_bias + (size_t)(l * 4 + 2) * 64,
        hsrc, csum, csq, NT);
    k_bn<<<(NT * 64 + 255) / 256, 256, 0, stream>>>(
        hsrc, csum, csq, bn_gamma + (size_t)(l * 2 + 1) * 64,
        bn_beta + (size_t)(l * 2 + 1) * 64, (l > 0) ? cxt : nullptr, oxt, NT);
    // agent-target relations (1, 3): src=track, dst=agent
    fill(acca, 0.f, NA * 256);
    do_rel(l, 1, cxt, NT, cxa, NA, acca);
    do_rel(l, 3, cxt, NT, cxa, NA, acca);
    fill(csum, 0.f, 64); fill(csq, 0.f, 64);
    k_finalize<<<(NA * 64 + 255) / 256, 256, 0, stream>>>(
        acca, conv_bias + (size_t)(l * 4 + 1) * 64, conv_bias + (size_t)(l * 4 + 3) * 64,
        hsrc, csum, csq, NA);
    k_bn<<<(NA * 64 + 255) / 256, 256, 0, stream>>>(
        hsrc, csum, csq, bn_gamma + (size_t)(l * 2 + 0) * 64,
        bn_beta + (size_t)(l * 2 + 0) * 64, (l > 0) ? cxa : nullptr, oxa, NA);
    float* t1 = cxa; cxa = oxa; oxa = t1;
    float* t2 = cxt; cxt = oxt; oxt = t2;
  }

  // ---- actor heads ----
  float* out = (float*)d_out;
  // agent alpha / beta
  gemm(cxa, aW1 + 0 * 64 * 32, ab1 + 0 * 32, h1b, NA, 64, 32, 0);
  k_head<<<(NA + 255) / 256, 256, 0, stream>>>(h1b, ag + 0, abe + 0, aW2 + 0, ab2 + 0, out + 0, NA);
  gemm(cxa, aW1 + 1 * 64 * 32, ab1 + 1 * 32, h1b, NA, 64, 32, 0);
  k_head<<<(NA + 255) / 256, 256, 0, stream>>>(h1b, ag + 32, abe + 32, aW2 + 32, ab2 + 1, out + NA, NA);
  k_ratio<<<(NA + 255) / 256, 256, 0, stream>>>(out, out + NA, out + 2 * NA, NA);
  // track alpha / beta
  float* tout = out + 3 * NA;
  gemm(cxt, aW1 + 2 * 64 * 32, ab1 + 2 * 32, h1b, NT, 64, 32, 0);
  k_head<<<(NT + 255) / 256, 256, 0, stream>>>(h1b, ag + 64, abe + 64, aW2 + 64, ab2 + 2, tout, NT);
  gemm(cxt, aW1 + 3 * 64 * 32, ab1 + 3 * 32, h1b, NT, 64, 32, 0);
  k_head<<<(NT + 255) / 256, 256, 0, stream>>>(h1b, ag + 96, abe + 96, aW2 + 96, ab2 + 3, tout + NT, NT);
  k_ratio<<<(NT + 255) / 256, 256, 0, stream>>>(tout, tout + NT, tout + 2 * NT, NT);

  // ---- critic global value ----
  fill(cstat, 0.f, 128);            // sum_a, sumsq_a
  fill(cstat + 128, -3e38f, 64);    // max_a
  fill(cstat + 192, 0.f, 128);      // sum_t, sumsq_t
  fill(cstat + 320, -3e38f, 64);    // max_t
  k_colstats<<<(NA * 64 + 255) / 256, 256, 0, stream>>>(cxa, cstat, cstat + 64, cstat + 128, NA);
  k_colstats<<<(NT * 64 + 255) / 256, 256, 0, stream>>>(cxt, cstat + 192, cstat + 256, cstat + 320, NT);
  k_critic<<<1, 128, 0, stream>>>(cstat, cstat + 64, cstat + 128,
                                  cstat + 192, cstat + 256, cstat + 320,
                                  cW1, cb1, cg1, cbe1, cW2, cb2, cg2, cbe2,
                                  cW3, cb3, cW4, cb4,
                                  out + 3 * NA + 3 * NT, NA, NT);
  (void)n_in; (void)out_size; (void)ws_size;
}